// SIREN_Criterion_4827543240809
// MI455X (gfx1250) — compile-verified
//
#include <hip/hip_runtime.h>
#include <hip/hip_bf16.h>
#include <math.h>

typedef __attribute__((ext_vector_type(2))) float v2f;
typedef __attribute__((ext_vector_type(8))) float v8f;

#define DIMS 32
#define NHLOG2PI  (-0.918938533204672741780329736406f)
#define LOG2E     (1.44269504088896340736f)

// ---------------------------------------------------------------------------
// K1: x_n = x / max(||x||, eps)   (one wave32 per row, D=32 == wave width)
// ---------------------------------------------------------------------------
__global__ void k_norm_rows(const float* __restrict__ x, float* __restrict__ xn,
                            int N, float eps) {
    int lane = threadIdx.x & 31;
    int row  = (int)((blockIdx.x * blockDim.x + threadIdx.x) >> 5);
    if (row >= N) return;
    float v  = x[(size_t)row * DIMS + lane];
    float ss = v * v;
    #pragma unroll
    for (int m = 1; m < 32; m <<= 1) ss += __shfl_xor(ss, m, 32);
    float nrm = fmaxf(sqrtf(ss), eps);
    xn[(size_t)row * DIMS + lane] = v / nrm;
}

// ---------------------------------------------------------------------------
// K2: sequential normalized-EMA scan, one thread per class (order-preserving).
//     Targets are streamed through LDS in 2048-entry chunks.
//     Final prototypes are normalized with eps 1e-8 and written as p_n.
// ---------------------------------------------------------------------------
__global__ void k_ema_scan(const float* __restrict__ x,       // raw id_samples [N,32]
                           const int*   __restrict__ targets, // [N]
                           const float* __restrict__ protos0, // [C,32]
                           float* __restrict__ pn,            // out [C,32]
                           int N, int C) {
    __shared__ int tgt[2048];
    int c = blockIdx.x * blockDim.x + threadIdx.x;

    float p[DIMS];
    if (c < C) {
        #pragma unroll
        for (int d = 0; d < DIMS; ++d) p[d] = protos0[(size_t)c * DIMS + d];
    }

    for (int base = 0; base < N; base += 2048) {
        int chunk = min(2048, N - base);
        for (int i = threadIdx.x; i < chunk; i += blockDim.x)
            tgt[i] = targets[base + i];
        __syncthreads();
        if (c < C) {
            for (int i = 0; i < chunk; ++i) {
                if (tgt[i] == c) {
                    const float* xr = x + (size_t)(base + i) * DIMS;
                    float s[DIMS];
                    float ss = 0.f;
                    #pragma unroll
                    for (int d = 0; d < DIMS; ++d) { s[d] = xr[d]; ss += s[d] * s[d]; }
                    float n1 = fmaxf(sqrtf(ss), 1e-12f);      // eps_n
                    float ss2 = 0.f;
                    #pragma unroll
                    for (int d = 0; d < DIMS; ++d) {
                        p[d] = 0.05f * (s[d] / n1) + 0.95f * p[d];
                        ss2 += p[d] * p[d];
                    }
                    float n2 = fmaxf(sqrtf(ss2), 1e-12f);     // eps_n
                    #pragma unroll
                    for (int d = 0; d < DIMS; ++d) p[d] = p[d] / n2;
                }
            }
        }
        __syncthreads();
    }

    if (c < C) {
        float ss = 0.f;
        #pragma unroll
        for (int d = 0; d < DIMS; ++d) ss += p[d] * p[d];
        float n = fmaxf(sqrtf(ss), 1e-8f);                    // eps_c
        #pragma unroll
        for (int d = 0; d < DIMS; ++d) pn[(size_t)c * DIMS + d] = p[d] / n;
    }
}

// ---------------------------------------------------------------------------
// K3: kap = relu(kw);  w = exp(logC_d(kap)) via uniform asymptotic Bessel.
//     Packs {kap, w} as float2 so K4 fetches both with one b64 load.
// ---------------------------------------------------------------------------
__global__ void k_kappa(const float* __restrict__ kw, float2* __restrict__ kapw,
                        int C, float Dv) {
    int c = blockIdx.x * blockDim.x + threadIdx.x;
    if (c >= C) return;
    float k = fmaxf(kw[c], 0.0f);
    float nu = 0.5f * Dv - 1.0f;                   // s = d/2 - 1
    float z  = k / nu;
    float zz = z * z;
    float sq = sqrtf(1.0f + zz);
    float eta = sq + logf(z) - log1pf(sq);
    float t  = 1.0f / sq;
    float u1 = (3.0f * t - 5.0f * t * t * t) / 24.0f;
    float logI = nu * eta - 0.5f * logf(2.0f * (float)M_PI * nu)
               - 0.25f * log1pf(zz) + log1pf(u1 / nu);
    float logC = Dv * NHLOG2PI + nu * logf(k) - logI;
    kapw[c] = make_float2(k, expf(logC));
}

// ---------------------------------------------------------------------------
// K4: fused  cos = x_n @ p_n^T  (WMMA f32 16x16x4, K=32)  ->  num = w*exp(k*cos)
//     -> rowsum over all C, capture num at target  ->  rowloss.
//     One wave per 32 rows (two 16-row M-tiles sharing each B load).
//     Grid exactly covers N, so EXEC is all-ones at every WMMA.
// Layouts (CDNA5 ISA 7.12.2, 32-bit A 16x4 / mirrored B 4x16 / C-D 16x16):
//   A lane l : { A[M=l&15][k0], A[M=l&15][k0+1] },  k0 = 4*chunk + 2*(l>>4)
//   B lane l : { B[k0][N=l&15], B[k0+1][N=l&15] }   (same k0)
//   D vgpr j, lane l : D[M = j + 8*(l>>4)][N = l&15]
// ---------------------------------------------------------------------------
__global__ void k_vmf_main(const float*  __restrict__ xn,    // [N,32]
                           const float*  __restrict__ pn,    // [C,32]
                           const float2* __restrict__ kapw,  // [C] {kap, w}
                           const int*    __restrict__ targets,
                           float* __restrict__ rowloss,      // [N]
                           int N, int C) {
    int lane = threadIdx.x & 31;
    int gw   = (int)((blockIdx.x * blockDim.x + threadIdx.x) >> 5);
    int rowBase = gw * 32;
    if (rowBase >= N) return;
    int half = lane >> 4;     // 0 or 1
    int lm   = lane & 15;

    // Two A tiles (rows rowBase..+15 and rowBase+16..+31), ISA striping.
    v2f a0[8], a1[8];
    const float* xr0 = xn + (size_t)(rowBase + lm) * DIMS;
    const float* xr1 = xn + (size_t)(rowBase + 16 + lm) * DIMS;
    #pragma unroll
    for (int ch = 0; ch < 8; ++ch) {
        int k0 = ch * 4 + half * 2;
        a0[ch] = *(const v2f*)(xr0 + k0);
        a1[ch] = *(const v2f*)(xr1 + k0);
    }

    // Targets of the 8+8 rows each half-lane-group owns.
    int tr0[8], tr1[8];
    #pragma unroll
    for (int j = 0; j < 8; ++j) {
        tr0[j] = targets[rowBase + half * 8 + j];
        tr1[j] = targets[rowBase + 16 + half * 8 + j];
    }

    float ps0[8], ps1[8], tn0[8], tn1[8];
    #pragma unroll
    for (int j = 0; j < 8; ++j) { ps0[j] = ps1[j] = tn0[j] = tn1[j] = 0.f; }

    for (int cBase = 0; cBase < C; cBase += 16) {
        int cls = cBase + lm;
        const float* prow = pn + (size_t)cls * DIMS;

        v8f acc0 = {}, acc1 = {};
        #pragma unroll
        for (int ch = 0; ch < 8; ++ch) {
            int k0 = ch * 4 + half * 2;
            v2f b = *(const v2f*)(prow + k0);      // shared by both M-tiles
            acc0 = __builtin_amdgcn_wmma_f32_16x16x4_f32(
                     false, a0[ch], false, b, (short)0, acc0, false, false);
            acc1 = __builtin_amdgcn_wmma_f32_16x16x4_f32(
                     false, a1[ch], false, b, (short)0, acc1, false, false);
        }

        float2 kwv = kapw[cls];                     // {kap_c, w_c}
        #pragma unroll
        for (int j = 0; j < 8; ++j) {
            // exp(k*cos) = exp2(log2e * k * cos): single v_exp_f32, |arg| < 1.5
            float n0 = kwv.y * __builtin_amdgcn_exp2f(LOG2E * kwv.x * acc0[j]);
            float n1 = kwv.y * __builtin_amdgcn_exp2f(LOG2E * kwv.x * acc1[j]);
            ps0[j] += n0;
            ps1[j] += n1;
            if (tr0[j] == cls) tn0[j] = n0;
            if (tr1[j] == cls) tn1[j] = n1;
        }
    }

    // Reduce across the 16 lanes of each half (masks 1..8 stay inside a half).
    #pragma unroll
    for (int j = 0; j < 8; ++j) {
        float s0 = ps0[j], t0 = tn0[j], s1 = ps1[j], t1 = tn1[j];
        #pragma unroll
        for (int m = 1; m < 16; m <<= 1) {
            s0 += __shfl_xor(s0, m, 32);
            t0 += __shfl_xor(t0, m, 32);
            s1 += __shfl_xor(s1, m, 32);
            t1 += __shfl_xor(t1, m, 32);
        }
        if (lm == 0) {
            rowloss[rowBase + half * 8 + j]      = -logf(t0 / s0 + 1e-6f);
            rowloss[rowBase + 16 + half * 8 + j] = -logf(t1 / s1 + 1e-6f);
        }
    }
}

// ---------------------------------------------------------------------------
// K5: deterministic single-block mean of rowloss -> scalar out.
// ---------------------------------------------------------------------------
__global__ void k_mean(const float* __restrict__ rowloss, float* __restrict__ out, int N) {
    __shared__ float sm[256];
    float acc = 0.f;
    for (int i = threadIdx.x; i < N; i += 256) acc += rowloss[i];
    sm[threadIdx.x] = acc;
    __syncthreads();
    #pragma unroll
    for (int s = 128; s > 0; s >>= 1) {
        if ((int)threadIdx.x < s) sm[threadIdx.x] += sm[threadIdx.x + s];
        __syncthreads();
    }
    if (threadIdx.x == 0) out[0] = sm[0] / (float)N;
}

// ---------------------------------------------------------------------------
extern "C" void kernel_launch(void* const* d_in, const int* in_sizes, int n_in,
                              void* d_out, int out_size, void* d_ws, size_t ws_size,
                              hipStream_t stream) {
    const float* id_samples   = (const float*)d_in[0]; // [N, D]
    const float* kappa_weight = (const float*)d_in[1]; // [C]
    const float* prototypes   = (const float*)d_in[2]; // [C, D]
    const int*   targets      = (const int*)  d_in[3]; // [N]

    const int C = in_sizes[1];
    const int N = in_sizes[3];
    const float Dv = (float)(in_sizes[0] / N);         // project_dim (== 32)

    // Workspace layout (floats)
    float*  ws      = (float*)d_ws;
    float*  xn      = ws;                              // N*32
    float*  pn      = xn + (size_t)N * DIMS;           // C*32
    float2* kapw    = (float2*)(pn + (size_t)C * DIMS);// C float2
    float*  rowloss = (float*)(kapw + C);              // N

    // K1: normalized samples for cosine (eps 1e-8).
    {
        int rowsPerBlock = 256 / 32;
        int blocks = (N + rowsPerBlock - 1) / rowsPerBlock;
        k_norm_rows<<<blocks, 256, 0, stream>>>(id_samples, xn, N, 1e-8f);
    }
    // K2: sequential EMA prototype scan + final normalize -> p_n.
    {
        int blocks = (C + 255) / 256;
        k_ema_scan<<<blocks, 256, 0, stream>>>(id_samples, targets, prototypes, pn, N, C);
    }
    // K3: kappa / vMF partition weight (packed float2).
    {
        int blocks = (C + 255) / 256;
        k_kappa<<<blocks, 256, 0, stream>>>(kappa_weight, kapw, C, Dv);
    }
    // K4: WMMA fused GEMM + weighted-softmax gather. One wave per 32 rows.
    {
        int waves  = N / 32;             // N = 16384 -> 512 waves
        int wavesPerBlock = 4;           // 128 threads
        int blocks = waves / wavesPerBlock;
        k_vmf_main<<<blocks, wavesPerBlock * 32, 0, stream>>>(
            xn, pn, kapw, targets, rowloss, N, C);
    }
    // K5: mean -> scalar.
    k_mean<<<1, 256, 0, stream>>>(rowloss, (float*)d_out, N);
}